// SubgraphRandomMasker_88450556494514
// MI455X (gfx1250) — compile-verified
//
#include <hip/hip_runtime.h>
#include <stdint.h>

// ---------------------------------------------------------------------------
// SubgraphRandomMasker for MI455X (gfx1250, wave32).
// Irregular bandwidth/atomics-bound pipeline: no matmul => no WMMA; we use
//  - CDNA5 async global->LDS (ASYNCcnt) to double-buffer the edge stream,
//  - the Tensor Data Mover (TENSORcnt) to stage contiguous edge tiles,
//  - wave32 ballot compaction + deterministic prefix-sum stream compaction.
// ---------------------------------------------------------------------------

#define NUM_ALL    200000
#define NUM_EDGES  4000000
#define NUM_SEEDS  2048
#define SAMPLE_NUM 20000            // NUM_ALL * KEEP_RATE

#define TILE            256
#define NUM_TILES       (NUM_EDGES / TILE)                    // 15625 (exact)
#define TILES_PER_BLOCK 4
#define MARK_BLOCKS     ((NUM_TILES + TILES_PER_BLOCK - 1) / TILES_PER_BLOCK)
#define NODE_BLOCKS     ((NUM_ALL + 255) / 256)               // 782

#define EMPTY_KEY 0xFFFFFFFFFFFFFFFFull
#define DEC_CAP   13000000LL        // max decoder edges we will emit

// byte offsets of output regions inside d_out (worst-case layout, all stores
// are range-guarded against the harness-provided out_size)
#define OFF_ENC_ROWS 0LL
#define OFF_ENC_COLS (8LL * NUM_EDGES)            //  32 MB
#define OFF_ENC_W    (16LL * NUM_EDGES)           //  64 MB
#define OFF_DEC_ROWS (20LL * NUM_EDGES)           //  80 MB
#define OFF_DEC_COLS (20LL * NUM_EDGES + 8LL * DEC_CAP)

#if __has_builtin(__builtin_amdgcn_tensor_load_to_lds)
#define HAVE_TDM 1
#else
#define HAVE_TDM 0
#endif

typedef __attribute__((ext_vector_type(4))) unsigned uint32x4;
typedef __attribute__((ext_vector_type(4))) int      int32x4;
typedef __attribute__((ext_vector_type(8))) int      int32x8;

// ---------------------------------------------------------------- utilities
static __device__ __forceinline__ unsigned ballot32(bool p) {
#if __has_builtin(__builtin_amdgcn_ballot_w32)
  return __builtin_amdgcn_ballot_w32(p);
#else
  return (unsigned)__ballot(p);
#endif
}

static __device__ __forceinline__ unsigned long long splitmix64(unsigned long long x) {
  x += 0x9E3779B97F4A7C15ull;
  x = (x ^ (x >> 30)) * 0xBF58476D1CE4E5B9ull;
  x = (x ^ (x >> 27)) * 0x94D049BB133111EBull;
  return x ^ (x >> 31);
}

// CDNA5 async global->LDS copy (tracked with ASYNCcnt). Each lane loads its
// own 8 bytes into its own LDS slot, so a per-wave s_wait_asynccnt suffices.
static __device__ __forceinline__ void async_load_b64_to_lds(void* lds, const void* gaddr) {
  unsigned l = (unsigned)(uintptr_t)lds;                 // low 32b = LDS byte addr
  unsigned long long g = (unsigned long long)(uintptr_t)gaddr;
  asm volatile("global_load_async_to_lds_b64 %0, %1, off" :: "v"(l), "v"(g) : "memory");
}

static __device__ __forceinline__ void wait_async(int keep_pending_tile) {
#if __has_builtin(__builtin_amdgcn_s_wait_asynccnt)
  if (keep_pending_tile) __builtin_amdgcn_s_wait_asynccnt(2);
  else                   __builtin_amdgcn_s_wait_asynccnt(0);
#else
  if (keep_pending_tile) asm volatile("s_wait_asynccnt 0x2" ::: "memory");
  else                   asm volatile("s_wait_asynccnt 0x0" ::: "memory");
#endif
}

static __device__ __forceinline__ void wait_tensor0() {
#if __has_builtin(__builtin_amdgcn_s_wait_tensorcnt)
  __builtin_amdgcn_s_wait_tensorcnt(0);
#else
  asm volatile("s_wait_tensorcnt 0x0" ::: "memory");
#endif
}

#if HAVE_TDM
// Tensor Data Mover: 1-D contiguous copy of nElem 8-byte elements into LDS.
// D# group0: count=1 | lds_addr | global_addr[56:0] | type=2("image")
// D# group1: data_size=3(8B), tensor_dim0=nElem, tensor_dim1=1,
//            tile_dim0=nElem (1-D tile), stride0=nElem. Groups 2/3 unused (<=2D).
static __device__ __forceinline__ void tdm_load_1d_b64(void* lds, const void* gaddr,
                                                       unsigned nElem) {
  const unsigned long long ga = (unsigned long long)(uintptr_t)gaddr;
  uint32x4 g0;
  g0.x = 1u;                                               // count=1, user mode
  g0.y = (unsigned)(uintptr_t)lds;                         // lds_addr (bytes)
  g0.z = (unsigned)(ga & 0xFFFFFFFFu);                     // global_addr[31:0]
  g0.w = (unsigned)((ga >> 32) & 0x01FFFFFFu) | (2u << 30);// addr[56:32] | type=2
  int32x8 g1;
  g1[0] = (int)(3u << 16);                                 // data_size = 8B
  g1[1] = (int)(nElem << 16);                              // tensor_dim0 lo16
  g1[2] = (int)((nElem >> 16) | (1u << 16));               // dim0 hi16 | dim1 lo16=1
  g1[3] = (int)(nElem << 16);                              // dim1 hi16=0 | tile_dim0
  g1[4] = 0;                                               // tile_dim1=0, tile_dim2=0
  g1[5] = (int)nElem;                                      // stride0 lo32
  g1[6] = 0;
  g1[7] = 0;
  const int32x4 z4 = {0, 0, 0, 0};
#if defined(__clang_major__) && (__clang_major__ >= 23)
  const int32x8 z8 = {0, 0, 0, 0, 0, 0, 0, 0};
  __builtin_amdgcn_tensor_load_to_lds(g0, g1, z4, z4, z8, 0);
#else
  __builtin_amdgcn_tensor_load_to_lds(g0, g1, z4, z4, 0);
#endif
}
#endif  // HAVE_TDM

static __device__ __forceinline__ void store_i64(char* out, long long byteoff,
                                                 long long cap, long long v) {
  if (byteoff + 8 <= cap) *(long long*)(out + byteoff) = v;
}
static __device__ __forceinline__ void store_f32(char* out, long long byteoff,
                                                 long long cap, float v) {
  if (byteoff + 4 <= cap) *(float*)(out + byteoff) = v;
}

static __device__ __forceinline__ void ht_insert(unsigned long long* tbl,
                                                 unsigned capMask,
                                                 unsigned long long key) {
  unsigned h = (unsigned)splitmix64(key) & capMask;
  for (unsigned probe = 0; probe <= capMask; ++probe) {
    unsigned long long prev = atomicCAS(&tbl[h], EMPTY_KEY, key);
    if (prev == EMPTY_KEY || prev == key) return;
    h = (h + 1) & capMask;
  }
}

// ----------------------------------------------------------------- kernels
__global__ void k_zero_u32(unsigned* __restrict__ p, long long n) {
  long long i = (long long)blockIdx.x * blockDim.x + threadIdx.x;
  long long stride = (long long)gridDim.x * blockDim.x;
  for (; i < n; i += stride) p[i] = 0u;
}

__global__ void k_fill_u64(unsigned long long* __restrict__ p, long long n,
                           unsigned long long v) {
  long long i = (long long)blockIdx.x * blockDim.x + threadIdx.x;
  long long stride = (long long)gridDim.x * blockDim.x;
  for (; i < n; i += stride) p[i] = v;
}

__global__ void k_scatter_seeds(const long long* __restrict__ seeds, int n,
                                unsigned* __restrict__ flags,
                                unsigned* __restrict__ maskflag) {
  int i = blockIdx.x * blockDim.x + threadIdx.x;
  if (i < n) {
    int s = (int)seeds[i];
    flags[s] = 1u;
    maskflag[s] = 1u;   // seeds are part of the mask node set
  }
}

__global__ void k_sample_nodes(unsigned* __restrict__ maskflag) {
  int i = blockIdx.x * blockDim.x + threadIdx.x;
  if (i < SAMPLE_NUM) {
    unsigned node = (unsigned)(splitmix64(0x5EED5EEDull + (unsigned long long)i) % NUM_ALL);
    maskflag[node] = 1u;
  }
}

// One BFS hop + degree accumulation, double-buffered via async LDS loads.
__global__ void __launch_bounds__(256)
k_mark_touched(const long long* __restrict__ rows, const long long* __restrict__ cols,
               const unsigned* __restrict__ flags, unsigned* __restrict__ maskflag,
               unsigned* __restrict__ deg, unsigned* __restrict__ keptBits,
               unsigned* __restrict__ tileCounts) {
  __shared__ long long sR[2][TILE];
  __shared__ long long sC[2][TILE];
  __shared__ unsigned wcnt[8];
  const int tid = threadIdx.x;
  const int t0 = blockIdx.x * TILES_PER_BLOCK;
  const int t1 = min(t0 + TILES_PER_BLOCK, NUM_TILES);
  const int nt = t1 - t0;
  if (nt <= 0) return;

  {  // prologue: issue tile 0 into buffer 0 (2 async ops per tile per wave)
    const long long e = (long long)t0 * TILE + tid;
    async_load_b64_to_lds(&sR[0][tid], rows + e);
    async_load_b64_to_lds(&sC[0][tid], cols + e);
  }

  for (int k = 0; k < nt; ++k) {
    const int buf = k & 1;
    if (k + 1 < nt) {  // prefetch next tile into the other buffer
      const int nb = (k + 1) & 1;
      const long long e = (long long)(t0 + k + 1) * TILE + tid;
      async_load_b64_to_lds(&sR[nb][tid], rows + e);
      async_load_b64_to_lds(&sC[nb][tid], cols + e);
      wait_async(1);   // allow next tile's 2 ops to stay in flight
    } else {
      wait_async(0);
    }

    const int tile = t0 + k;
    const int r = (int)sR[buf][tid];   // ds_load_b64
    const int c = (int)sC[buf][tid];

    const unsigned fr = flags[r];
    const unsigned fc = flags[c];
    const bool touched = (fr | fc) != 0u;

    const unsigned kb = ballot32(!touched);
    const unsigned lane = (unsigned)tid & 31u;
    const unsigned wave = (unsigned)tid >> 5;
    if (lane == 0u) {
      keptBits[tile * 8 + wave] = kb;
      wcnt[wave] = (unsigned)__popc(kb);
    }
    if (touched) {  // endpoints of touched edges join the mask node set
      maskflag[r] = 1u;
      maskflag[c] = 1u;
    } else {        // kept edge: unit-weight degree over dst (col)
      atomicAdd(&deg[c], 1u);
    }
    __syncthreads();
    if (tid == 0) {
      unsigned s = 0;
#pragma unroll
      for (int w = 0; w < 8; ++w) s += wcnt[w];
      tileCounts[tile] = s;
    }
    __syncthreads();
  }
}

// Single-block exclusive scan (deterministic compaction offsets).
__global__ void __launch_bounds__(256)
k_scan(const unsigned* __restrict__ counts, unsigned* __restrict__ offsets,
       int n, unsigned* __restrict__ total) {
  __shared__ unsigned s[256];
  __shared__ unsigned carry;
  if (threadIdx.x == 0) carry = 0u;
  __syncthreads();
  for (int base = 0; base < n; base += 256) {
    const int i = base + (int)threadIdx.x;
    const unsigned v = (i < n) ? counts[i] : 0u;
    s[threadIdx.x] = v;
    __syncthreads();
    for (int d = 1; d < 256; d <<= 1) {
      const unsigned t = (threadIdx.x >= (unsigned)d) ? s[threadIdx.x - d] : 0u;
      __syncthreads();
      s[threadIdx.x] += t;
      __syncthreads();
    }
    const unsigned incl = s[threadIdx.x];
    if (i < n) offsets[i] = carry + (incl - v);
    __syncthreads();
    if (threadIdx.x == 255) carry += incl;
    __syncthreads();
  }
  if (threadIdx.x == 0 && total) *total = carry;
}

__global__ void k_dinv(const unsigned* __restrict__ deg, float* __restrict__ dinv) {
  int i = blockIdx.x * blockDim.x + threadIdx.x;
  if (i < NUM_ALL) {
    unsigned d = deg[i];
    dinv[i] = d ? rsqrtf((float)d) : 0.0f;
  }
}

// Deterministic encoder compaction. The contiguous row/col tile is staged
// through the Tensor Data Mover: one wave issues two TENSOR_LOAD_TO_LDS ops,
// waits on TENSORcnt, and the workgroup barrier publishes the tile.
__global__ void __launch_bounds__(256)
k_compact_edges(const long long* __restrict__ rows, const long long* __restrict__ cols,
                const unsigned* __restrict__ keptBits,
                const unsigned* __restrict__ tileOffsets,
                const float* __restrict__ dinv,
                char* __restrict__ out, long long cap) {
  const int tile = blockIdx.x;
  const int tid = threadIdx.x;
#if HAVE_TDM
  __shared__ long long sR[TILE];
  __shared__ long long sC[TILE];
  if (tid < 32) {   // wave 0 drives the TDM (EXEC ignored by tensor ops)
    tdm_load_1d_b64(sR, rows + (long long)tile * TILE, TILE);
    tdm_load_1d_b64(sC, cols + (long long)tile * TILE, TILE);
    wait_tensor0();
  }
  __syncthreads();
#endif
  const unsigned lane = (unsigned)tid & 31u;
  const unsigned wave = (unsigned)tid >> 5;
  const unsigned word = keptBits[tile * 8 + wave];
  const bool kept = (word >> lane) & 1u;
  unsigned base = 0;
  for (unsigned w = 0; w < wave; ++w) base += (unsigned)__popc(keptBits[tile * 8 + w]);
  const unsigned rank = (unsigned)__popc(word & ((1u << lane) - 1u));
  if (kept) {
#if HAVE_TDM
    const long long r = sR[tid];
    const long long c = sC[tid];
#else
    const int e = tile * TILE + tid;
    const long long r = rows[e];
    const long long c = cols[e];
#endif
    const unsigned o = tileOffsets[tile] + base + rank;
    const float wgt = dinv[(int)r] * dinv[(int)c];   // gcn_norm, unit weights
    store_i64(out, OFF_ENC_ROWS + 8LL * o, cap, r);
    store_i64(out, OFF_ENC_COLS + 8LL * o, cap, c);
    store_f32(out, OFF_ENC_W + 4LL * o, cap, wgt);
  }
}

__global__ void __launch_bounds__(256)
k_node_count(const unsigned* __restrict__ maskflag, unsigned* __restrict__ nodeCounts) {
  __shared__ unsigned wcnt[8];
  const int i = blockIdx.x * 256 + (int)threadIdx.x;
  const bool on = (i < NUM_ALL) && (maskflag[i] != 0u);
  const unsigned bal = ballot32(on);
  const unsigned lane = threadIdx.x & 31u;
  const unsigned wave = threadIdx.x >> 5;
  if (lane == 0u) wcnt[wave] = (unsigned)__popc(bal);
  __syncthreads();
  if (threadIdx.x == 0) {
    unsigned s = 0;
#pragma unroll
    for (int w = 0; w < 8; ++w) s += wcnt[w];
    nodeCounts[blockIdx.x] = s;
  }
}

__global__ void __launch_bounds__(256)
k_node_compact(const unsigned* __restrict__ maskflag,
               const unsigned* __restrict__ nodeOffsets,
               unsigned* __restrict__ maskList) {
  __shared__ unsigned wcnt[8];
  const int i = blockIdx.x * 256 + (int)threadIdx.x;
  const bool on = (i < NUM_ALL) && (maskflag[i] != 0u);
  const unsigned bal = ballot32(on);
  const unsigned lane = threadIdx.x & 31u;
  const unsigned wave = threadIdx.x >> 5;
  if (lane == 0u) wcnt[wave] = (unsigned)__popc(bal);
  __syncthreads();
  unsigned base = 0;
  for (unsigned w = 0; w < wave; ++w) base += wcnt[w];
  const unsigned rank = (unsigned)__popc(bal & ((1u << lane) - 1u));
  if (on) maskList[nodeOffsets[blockIdx.x] + base + rank] = (unsigned)i;
}

__global__ void k_insert_loops(unsigned long long* __restrict__ tbl, unsigned capMask) {
  const int i = blockIdx.x * blockDim.x + (int)threadIdx.x;
  if (i < NUM_ALL)
    ht_insert(tbl, capMask, (unsigned long long)i * NUM_ALL + (unsigned long long)i);
}

__global__ void __launch_bounds__(256)
k_insert_kept(const long long* __restrict__ rows, const long long* __restrict__ cols,
              const unsigned* __restrict__ keptBits,
              unsigned long long* __restrict__ tbl, unsigned capMask) {
  const int e = blockIdx.x * 256 + (int)threadIdx.x;   // grid covers exactly NUM_EDGES
  __builtin_prefetch(rows + e + 4096, 0, 0);           // global_prefetch_b8
  __builtin_prefetch(cols + e + 4096, 0, 0);
  const unsigned word = keptBits[e >> 5];
  if ((word >> ((unsigned)e & 31u)) & 1u) {
    const unsigned long long r = (unsigned long long)rows[e];
    const unsigned long long c = (unsigned long long)cols[e];
    ht_insert(tbl, capMask, r * NUM_ALL + c);
  }
}

__global__ void __launch_bounds__(256)
k_insert_random(const unsigned* __restrict__ maskList,
                const unsigned* __restrict__ counters,  // [1] = mask node count
                unsigned long long* __restrict__ tbl, unsigned capMask) {
  const int e = blockIdx.x * 256 + (int)threadIdx.x;
  const unsigned M = counters[1];
  if (M == 0u) return;
  const unsigned long long x = splitmix64(0x00C0FFEEull + (unsigned long long)e);
  const unsigned long long y = splitmix64(0xDEADBEEFull + (unsigned long long)e);
  const unsigned long long r = (unsigned long long)maskList[(unsigned)(x % M)];
  const unsigned long long c = (unsigned long long)maskList[(unsigned)(y % M)];
  ht_insert(tbl, capMask, r * NUM_ALL + c);   // tem_rows->tem_cols
  ht_insert(tbl, capMask, c * NUM_ALL + r);   // and the symmetric pair
}

__global__ void __launch_bounds__(256)
k_tbl_count(const unsigned long long* __restrict__ tbl, unsigned* __restrict__ tblCounts) {
  __shared__ unsigned wcnt[8];
  const long long i = (long long)blockIdx.x * 256 + threadIdx.x;
  const bool occ = tbl[i] != EMPTY_KEY;
  const unsigned bal = ballot32(occ);
  const unsigned lane = threadIdx.x & 31u;
  const unsigned wave = threadIdx.x >> 5;
  if (lane == 0u) wcnt[wave] = (unsigned)__popc(bal);
  __syncthreads();
  if (threadIdx.x == 0) {
    unsigned s = 0;
#pragma unroll
    for (int w = 0; w < 8; ++w) s += wcnt[w];
    tblCounts[blockIdx.x] = s;
  }
}

__global__ void __launch_bounds__(256)
k_tbl_compact(const unsigned long long* __restrict__ tbl,
              const unsigned* __restrict__ tblOffsets,
              char* __restrict__ out, long long cap) {
  __shared__ unsigned wcnt[8];
  const long long i = (long long)blockIdx.x * 256 + threadIdx.x;
  const unsigned long long key = tbl[i];
  const bool occ = key != EMPTY_KEY;
  const unsigned bal = ballot32(occ);
  const unsigned lane = threadIdx.x & 31u;
  const unsigned wave = threadIdx.x >> 5;
  if (lane == 0u) wcnt[wave] = (unsigned)__popc(bal);
  __syncthreads();
  unsigned base = 0;
  for (unsigned w = 0; w < wave; ++w) base += wcnt[w];
  const unsigned rank = (unsigned)__popc(bal & ((1u << lane) - 1u));
  if (occ) {
    const unsigned o = tblOffsets[blockIdx.x] + base + rank;
    if ((long long)o < DEC_CAP) {
      const long long dc = (long long)(key % NUM_ALL);
      const long long dr = (long long)(key / NUM_ALL);
      store_i64(out, OFF_DEC_ROWS + 8LL * o, cap, dr);
      store_i64(out, OFF_DEC_COLS + 8LL * o, cap, dc);
    }
  }
}

// ------------------------------------------------------------------- launch
extern "C" void kernel_launch(void* const* d_in, const int* in_sizes, int n_in,
                              void* d_out, int out_size, void* d_ws, size_t ws_size,
                              hipStream_t stream) {
  (void)in_sizes; (void)n_in;
  const long long* edge_index = (const long long*)d_in[0];  // int64 [2, NUM_EDGES]
  const long long* seeds      = (const long long*)d_in[2];  // int64 [NUM_SEEDS]
  const long long* rows = edge_index;
  const long long* cols = edge_index + NUM_EDGES;
  char* out = (char*)d_out;
  const long long out_cap = (long long)out_size * 4;        // bytes

  // ---- carve workspace (256B aligned regions)
  uint8_t* base = (uint8_t*)d_ws;
  size_t off = 0;
  auto carve = [&](size_t bytes) -> uint8_t* {
    uint8_t* p = base + off;
    off += (bytes + 255) & ~(size_t)255;
    return p;
  };
  unsigned* flags       = (unsigned*)carve(4ull * NUM_ALL);
  unsigned* maskflag    = (unsigned*)carve(4ull * NUM_ALL);
  unsigned* deg         = (unsigned*)carve(4ull * NUM_ALL);
  float*    dinv        = (float*)   carve(4ull * NUM_ALL);
  unsigned* keptBits    = (unsigned*)carve(4ull * (NUM_EDGES / 32));
  unsigned* tileCounts  = (unsigned*)carve(4ull * NUM_TILES);
  unsigned* tileOffsets = (unsigned*)carve(4ull * NUM_TILES);
  unsigned* nodeCounts  = (unsigned*)carve(4ull * NODE_BLOCKS);
  unsigned* nodeOffsets = (unsigned*)carve(4ull * NODE_BLOCKS);
  unsigned* maskList    = (unsigned*)carve(4ull * NUM_ALL);
  unsigned* counters    = (unsigned*)carve(64);
  if (off > ws_size) return;  // not enough scratch: do nothing (deterministic)

  // hash table for decoder dedup: largest power-of-two that fits
  size_t cap = 0;
  for (size_t c = (size_t)1 << 25; c >= (size_t)1 << 20; c >>= 1) {
    const size_t need = c * 8 + 2 * ((c / 256) * 4) + 1024;
    if (off + need <= ws_size) { cap = c; break; }
  }
  unsigned*           tblCounts  = nullptr;
  unsigned*           tblOffsets = nullptr;
  unsigned long long* table      = nullptr;
  if (cap) {
    tblCounts  = (unsigned*)carve((cap / 256) * 4);
    tblOffsets = (unsigned*)carve((cap / 256) * 4);
    table      = (unsigned long long*)carve(cap * 8);
  }

  const dim3 B(256);
  // ---- init
  k_zero_u32<<<dim3(2048), B, 0, stream>>>(flags, 3LL * NUM_ALL);      // flags..deg contiguous
  k_zero_u32<<<dim3(1), dim3(64), 0, stream>>>(counters, 16);
  k_zero_u32<<<dim3(2048), B, 0, stream>>>((unsigned*)d_out, (long long)out_size);

  // ---- mask construction + one BFS hop
  k_scatter_seeds<<<dim3((NUM_SEEDS + 255) / 256), B, 0, stream>>>(seeds, NUM_SEEDS, flags, maskflag);
  k_sample_nodes<<<dim3((SAMPLE_NUM + 255) / 256), B, 0, stream>>>(maskflag);
  k_mark_touched<<<dim3(MARK_BLOCKS), B, 0, stream>>>(rows, cols, flags, maskflag, deg,
                                                      keptBits, tileCounts);

  // ---- encoder: deterministic compaction + gcn_norm
  k_scan<<<dim3(1), B, 0, stream>>>(tileCounts, tileOffsets, NUM_TILES, &counters[0]);
  k_dinv<<<dim3(NODE_BLOCKS), B, 0, stream>>>(deg, dinv);
  k_compact_edges<<<dim3(NUM_TILES), B, 0, stream>>>(rows, cols, keptBits, tileOffsets,
                                                     dinv, out, out_cap);

  // ---- mask node list
  k_node_count<<<dim3(NODE_BLOCKS), B, 0, stream>>>(maskflag, nodeCounts);
  k_scan<<<dim3(1), B, 0, stream>>>(nodeCounts, nodeOffsets, NODE_BLOCKS, &counters[1]);
  k_node_compact<<<dim3(NODE_BLOCKS), B, 0, stream>>>(maskflag, nodeOffsets, maskList);

  // ---- decoder: hash dedup of {random mask pairs, sym pairs, loops, kept edges}
  if (cap) {
    const unsigned capMask = (unsigned)(cap - 1);
    const int tblBlocks = (int)(cap / 256);
    k_fill_u64<<<dim3(4096), B, 0, stream>>>(table, (long long)cap, EMPTY_KEY);
    k_insert_loops<<<dim3(NODE_BLOCKS), B, 0, stream>>>(table, capMask);
    k_insert_kept<<<dim3(NUM_TILES), B, 0, stream>>>(rows, cols, keptBits, table, capMask);
    k_insert_random<<<dim3(NUM_TILES), B, 0, stream>>>(maskList, counters, table, capMask);
    k_tbl_count<<<dim3(tblBlocks), B, 0, stream>>>(table, tblCounts);
    k_scan<<<dim3(1), B, 0, stream>>>(tblCounts, tblOffsets, tblBlocks, &counters[2]);
    k_tbl_compact<<<dim3(tblBlocks), B, 0, stream>>>(table, tblOffsets, out, out_cap);
  }
}